// GNNModel_30451318129175
// MI455X (gfx1250) — compile-verified
//
#include <hip/hip_runtime.h>
#include <hip/hip_bf16.h>

typedef float v2f __attribute__((ext_vector_type(2)));
typedef float v8f __attribute__((ext_vector_type(8)));

#define NN 50000
#define EE 800000
#define FF 64
#define GG 1024

// ---------------------------------------------------------------------------
// WMMA f32 GEMM: C(M x Nc) = [relu]( A(M x K) @ W(K x Nc) [+ bias] )
// One wave (32 threads) per 16x16 output tile, accumulating with
// V_WMMA_F32_16X16X4_F32 over K in steps of 4.
// Layouts per CDNA5 ISA 7.12.2 (32-bit matrices, wave32):
//   A 16x4 : lane<16 -> K=0/1 in v[0]/v[1]; lane>=16 -> K=2/3. M = lane%16.
//   B 4x16 : lane<16 -> K=0/1 rows;        lane>=16 -> K=2/3. N = lane%16.
//   C 16x16: VGPR r holds M=r (lanes 0-15) and M=r+8 (lanes 16-31), N=lane%16.
// ---------------------------------------------------------------------------
template<bool BIAS, bool RELU>
__global__ __launch_bounds__(32)
void gemm_f32_wmma(const float* __restrict__ A, const float* __restrict__ W,
                   const float* __restrict__ bias, float* __restrict__ C,
                   int M, int K, int Nc) {
  const int lane   = threadIdx.x & 31;
  const int ntiles = Nc >> 4;
  const int mt     = blockIdx.x / ntiles;
  const int nt     = blockIdx.x % ntiles;

  const int m      = mt * 16 + (lane & 15);   // A row for this lane
  const int n      = nt * 16 + (lane & 15);   // B/C column for this lane
  const int khalf  = (lane >> 4) << 1;        // 0 for lanes 0-15, 2 for 16-31

  v8f acc = {};
  const float* __restrict__ arow = A + (size_t)m * K;
  for (int k = 0; k < K; k += 4) {
    v2f a, b;
    a.x = arow[k + khalf];
    a.y = arow[k + khalf + 1];
    b.x = W[(size_t)(k + khalf)     * Nc + n];
    b.y = W[(size_t)(k + khalf + 1) * Nc + n];
    acc = __builtin_amdgcn_wmma_f32_16x16x4_f32(
        /*neg_a=*/false, a, /*neg_b=*/false, b,
        /*c_mod=*/(short)0, acc, /*reuse_a=*/false, /*reuse_b=*/false);
  }

  const int mbase = mt * 16 + ((lane >> 4) << 3);  // +8 rows for upper half-wave
  float bv = BIAS ? bias[n] : 0.0f;
#pragma unroll
  for (int r = 0; r < 8; ++r) {
    float v = acc[r];
    if (BIAS) v += bv;
    if (RELU) v = fmaxf(v, 0.0f);
    C[(size_t)(mbase + r) * Nc + n] = v;
  }
}

// Degree: deg[dst[e]] += 1
__global__ void deg_kernel(const long long* __restrict__ dst, float* __restrict__ deg, int E) {
  int e = blockIdx.x * blockDim.x + threadIdx.x;
  if (e < E) atomicAdd(&deg[(int)dst[e]], 1.0f);
}

// dinv = rsqrt(deg + 1)
__global__ void dinv_kernel(const float* __restrict__ deg, float* __restrict__ dinv, int n) {
  int i = blockIdx.x * blockDim.x + threadIdx.x;
  if (i < n) dinv[i] = rsqrtf(deg[i] + 1.0f);
}

// acc = hW * dinv^2 + b   (self-loop term + bias; fully initializes acc)
__global__ void self_init_kernel(const float* __restrict__ hW, const float* __restrict__ dinv,
                                 const float* __restrict__ b, float* __restrict__ acc, int total) {
  int idx = blockIdx.x * blockDim.x + threadIdx.x;
  if (idx >= total) return;
  int i = idx >> 6, f = idx & 63;
  float d = dinv[i];
  acc[idx] = hW[idx] * d * d + b[f];
}

// Wave-per-edge scatter: acc[dst] += hW[src] * dinv[src]*dinv[dst]
// Lane L handles features L and L+32 (two contiguous 128B bursts per edge).
__global__ void edge_kernel(const long long* __restrict__ src, const long long* __restrict__ dst,
                            const float* __restrict__ hW, const float* __restrict__ dinv,
                            float* __restrict__ acc, int E) {
  const int lane   = threadIdx.x & 31;
  const int wid    = blockIdx.x * (blockDim.x >> 5) + (threadIdx.x >> 5);
  const int nwaves = gridDim.x * (blockDim.x >> 5);
  for (int e = wid; e < E; e += nwaves) {
    int s = (int)src[e], d = (int)dst[e];
    float norm = dinv[s] * dinv[d];
    const float* __restrict__ hs = hW + (size_t)s * 64;
    float*       __restrict__ ad = acc + (size_t)d * 64;
    atomicAdd(&ad[lane],      hs[lane]      * norm);
    atomicAdd(&ad[lane + 32], hs[lane + 32] * norm);
  }
}

// dst = relu(src)
__global__ void relu_kernel(const float* __restrict__ src, float* __restrict__ dst, int total) {
  int i = blockIdx.x * blockDim.x + threadIdx.x;
  if (i < total) dst[i] = fmaxf(src[i], 0.0f);
}

// Wave-per-node pooling: g[batch[i]] += h[i]
__global__ void pool_kernel(const long long* __restrict__ batch, const float* __restrict__ h,
                            float* __restrict__ g, int n) {
  const int lane   = threadIdx.x & 31;
  const int wid    = blockIdx.x * (blockDim.x >> 5) + (threadIdx.x >> 5);
  const int nwaves = gridDim.x * (blockDim.x >> 5);
  for (int i = wid; i < n; i += nwaves) {
    int bg = (int)batch[i];
    const float* __restrict__ hi = h + (size_t)i * 64;
    float*       __restrict__ gb = g + (size_t)bg * 64;
    atomicAdd(&gb[lane],      hi[lane]);
    atomicAdd(&gb[lane + 32], hi[lane + 32]);
  }
}

// coeff = g2 @ Wa + ba ; out = ((A - B/(T+C)) - mean)/std
__global__ void head_final_kernel(const float* __restrict__ g2, const float* __restrict__ Wa,
                                  const float* __restrict__ ba, const float* __restrict__ temp,
                                  const float* __restrict__ mean, const float* __restrict__ stdv,
                                  float* __restrict__ out, int G) {
  int gi = blockIdx.x * blockDim.x + threadIdx.x;
  if (gi >= G) return;
  float c0 = ba[0], c1 = ba[1], c2 = ba[2];
  const float* __restrict__ gv = g2 + (size_t)gi * 128;
#pragma unroll 4
  for (int k = 0; k < 128; ++k) {
    float v = gv[k];
    c0 += v * Wa[k * 3 + 0];
    c1 += v * Wa[k * 3 + 1];
    c2 += v * Wa[k * 3 + 2];
  }
  float logP = c0 - c1 / (temp[gi] + c2);
  out[gi] = (logP - mean[0]) / stdv[0];
}

extern "C" void kernel_launch(void* const* d_in, const int* in_sizes, int n_in,
                              void* d_out, int out_size, void* d_ws, size_t ws_size,
                              hipStream_t stream) {
  // Inputs (setup_inputs dict order):
  const float*     x     = (const float*)d_in[0];               // N x 64 f32
  const long long* ei    = (const long long*)d_in[1];           // 2 x E i64
  const long long* batch = (const long long*)d_in[2];           // N i64
  const float*     temp  = (const float*)d_in[3];               // G f32
  const float*     mean  = (const float*)d_in[4];               // scalar
  const float*     stdv  = (const float*)d_in[5];               // scalar
  const float* Wg[3] = { (const float*)d_in[6],  (const float*)d_in[8],  (const float*)d_in[10] };
  const float* bg[3] = { (const float*)d_in[7],  (const float*)d_in[9],  (const float*)d_in[11] };
  const float* Wd1 = (const float*)d_in[12]; const float* bd1 = (const float*)d_in[13];
  const float* Wd2 = (const float*)d_in[14]; const float* bd2 = (const float*)d_in[15];
  const float* Wa  = (const float*)d_in[16]; const float* ba  = (const float*)d_in[17];

  const int E = in_sizes[1] / 2;
  const long long* src = ei;
  const long long* dst = ei + E;

  // Workspace layout (floats)
  float* ws   = (float*)d_ws;
  float* deg  = ws;                    // N
  float* dinv = deg  + NN;             // N
  float* hW   = dinv + NN;             // N*64
  float* acc  = hW   + (size_t)NN * 64;// N*64
  float* h    = acc  + (size_t)NN * 64;// N*64
  float* g0   = h    + (size_t)NN * 64;// G*64
  float* g1   = g0   + (size_t)GG * 64;// G*256
  float* g2   = g1   + (size_t)GG * 256;// G*128

  // --- degree / normalization ---
  hipMemsetAsync(deg, 0, (size_t)NN * sizeof(float), stream);
  deg_kernel<<<(E + 255) / 256, 256, 0, stream>>>(dst, deg, E);
  dinv_kernel<<<(NN + 255) / 256, 256, 0, stream>>>(deg, dinv, NN);

  // --- 3 GCN layers ---
  const float* cur = x;
  for (int l = 0; l < 3; ++l) {
    // hW = cur @ Wg[l]   (WMMA f32)
    gemm_f32_wmma<false, false><<<(NN / 16) * (FF / 16), 32, 0, stream>>>(
        cur, Wg[l], nullptr, hW, NN, FF, FF);
    // acc = hW*dinv^2 + b
    self_init_kernel<<<((NN * FF) + 255) / 256, 256, 0, stream>>>(hW, dinv, bg[l], acc, NN * FF);
    // acc[dst] += hW[src]*norm
    edge_kernel<<<4096, 256, 0, stream>>>(src, dst, hW, dinv, acc, E);
    // h = relu(acc)
    relu_kernel<<<((NN * FF) + 255) / 256, 256, 0, stream>>>(acc, h, NN * FF);
    cur = h;
  }

  // --- pooling: g0 = relu(segment_sum(h, batch)) ---
  hipMemsetAsync(g0, 0, (size_t)GG * 64 * sizeof(float), stream);
  pool_kernel<<<2048, 256, 0, stream>>>(batch, h, g0, NN);
  relu_kernel<<<((GG * 64) + 255) / 256, 256, 0, stream>>>(g0, g0, GG * 64);

  // --- MLP head (WMMA f32 with fused bias+relu) ---
  gemm_f32_wmma<true, true><<<(GG / 16) * (256 / 16), 32, 0, stream>>>(
      g0, Wd1, bd1, g1, GG, 64, 256);
  gemm_f32_wmma<true, true><<<(GG / 16) * (128 / 16), 32, 0, stream>>>(
      g1, Wd2, bd2, g2, GG, 256, 128);

  // --- final 128->3 + Antoine equation ---
  head_final_kernel<<<(GG + 255) / 256, 256, 0, stream>>>(
      g2, Wa, ba, temp, mean, stdv, (float*)d_out, GG);
}